// RNNDecoder_75728863363147
// MI455X (gfx1250) — compile-verified
//
#include <hip/hip_runtime.h>
#include <hip/hip_bf16.h>
#include <stdint.h>

#define B_  128
#define L_  500
#define LP_ 512   // padded L (32 uniform tiles of 16)
#define H_  512
#define LD_ 256
#define UD_ 128
#define V_  33

typedef float v2f __attribute__((ext_vector_type(2)));
typedef float v8f __attribute__((ext_vector_type(8)));

__device__ inline v8f wmma4(v2f a, v2f b, v8f c) {
  // D = A(16x4 f32) * B(4x16 f32) + C(16x16 f32)
  return __builtin_amdgcn_wmma_f32_16x16x4_f32(false, a, false, b, (short)0, c,
                                               false, false);
}

#if __has_builtin(__builtin_amdgcn_s_wait_asynccnt)
#define WAIT_ASYNC(n) __builtin_amdgcn_s_wait_asynccnt(n)
#else
#define WAIT_ASYNC(n) asm volatile("s_wait_asynccnt %0" ::"i"(n) : "memory")
#endif

// Async DMA: 16 bytes per lane, global -> LDS (ASYNCcnt-tracked).
// lds_off is the LDS byte address (low 32 bits of a generic __shared__ pointer).
__device__ inline void async_load_b128(uint32_t lds_off, const void* gaddr) {
  asm volatile("global_load_async_to_lds_b128 %0, %1, off"
               :: "v"(lds_off), "v"(gaddr) : "memory");
}

// One wave computes a 16(M) x 64(N) tile of  C = A * W^T
//   a_row : per-lane pointer to row (mbase + (lane&15)) of A
//   W     : row-major [Ntotal, K]  (so B[k][n] = W[n*K + k])
__device__ inline void wave_tile(const float* __restrict__ a_row,
                                 const float* __restrict__ W, int K,
                                 int colbase, int lane, v8f acc[4]) {
  const int half = lane >> 4;
  const int nl   = lane & 15;
  for (int k0 = 0; k0 < K; k0 += 4) {
    v2f a = *(const v2f*)(a_row + k0 + 2 * half);
#pragma unroll
    for (int nn = 0; nn < 4; ++nn) {
      const int col = colbase + nn * 16 + nl;
      v2f bb = *(const v2f*)(W + (size_t)col * K + k0 + 2 * half);
      acc[nn] = wmma4(a, bb, acc[nn]);
    }
  }
}

// ---------------------------------------------------------------------------
// K0a: mem[b,l,h] = sum_u up[b,l,u] * mem_W[h,u] + mem_b[h]
// Output padded to LP_=512 rows per b. grid = B*32 blocks x 256 thr (8 waves);
// block covers one 16-row M tile of one b across all N=512.
// ---------------------------------------------------------------------------
__global__ void k_mem(const float* __restrict__ up, const float* __restrict__ mem_W,
                      const float* __restrict__ mem_b, float* __restrict__ mem) {
  const int lane = threadIdx.x & 31, wave = threadIdx.x >> 5;
  const int b  = blockIdx.x >> 5;       // 0..127
  const int mt = blockIdx.x & 31;       // 0..31 (16-row tiles of padded L)
  const int colbase = wave * 64;        // 8 waves cover N = 0..511
  const int l_row = mt * 16 + (lane & 15);
  const int l_src = (l_row < L_) ? l_row : (L_ - 1);   // clamp A gather for pad rows
  const float* a_row = up + ((size_t)b * L_ + l_src) * UD_;
  v8f acc[4] = {};
  wave_tile(a_row, mem_W, UD_, colbase, lane, acc);
  const int half = lane >> 4, nl = lane & 15;
#pragma unroll
  for (int nn = 0; nn < 4; ++nn) {
    const int n = colbase + nn * 16 + nl;
    const float bias = mem_b[n];
#pragma unroll
    for (int j = 0; j < 8; ++j) {
      const int l_dst = mt * 16 + j + 8 * half;        // pad rows get (masked) junk
      mem[((size_t)b * LP_ + l_dst) * H_ + n] = acc[nn][j] + bias;
    }
  }
}

// ---------------------------------------------------------------------------
// K0b: h0 = latent @ hid_W^T + hid_b      (M=128, N=512, K=256)
// grid = 8 blocks x 256 thr. Also initializes tok[b] = SOS(0).
// ---------------------------------------------------------------------------
__global__ void k_h0(const float* __restrict__ latent, const float* __restrict__ hid_W,
                     const float* __restrict__ hid_b, float* __restrict__ h,
                     int* __restrict__ tok) {
  if (blockIdx.x == 0 && threadIdx.x < B_) tok[threadIdx.x] = 0;  // SOS
  const int lane = threadIdx.x & 31, wave = threadIdx.x >> 5;
  const int mi = blockIdx.x;
  const int colbase = wave * 64;
  const int row = mi * 16 + (lane & 15);
  const float* a_row = latent + (size_t)row * LD_;
  v8f acc[4] = {};
  wave_tile(a_row, hid_W, LD_, colbase, lane, acc);
  const int half = lane >> 4, nl = lane & 15;
#pragma unroll
  for (int nn = 0; nn < 4; ++nn) {
    const int n = colbase + nn * 16 + nl;
    const float bias = hid_b[n];
#pragma unroll
    for (int j = 0; j < 8; ++j)
      h[(size_t)(mi * 16 + j + 8 * half) * H_ + n] = acc[nn][j] + bias;
  }
}

// ---------------------------------------------------------------------------
// K1: GI = embed[tok] @ W_ih^T ; GH = h @ W_hh^T    (M=128, N=1536, K=512 x2)
// grid = 48 blocks x 256 thr -> 384 waves; the 8 waves of a block share one
// N-chunk (same weight columns -> broadcast-friendly in WGP$/L2).
// ---------------------------------------------------------------------------
__global__ void k_gates_gemm(const float* __restrict__ embed, const float* __restrict__ h,
                             const int* __restrict__ tok,
                             const float* __restrict__ W_ih, const float* __restrict__ W_hh,
                             float* __restrict__ GI, float* __restrict__ GH) {
  const int lane = threadIdx.x & 31, wave = threadIdx.x >> 5;
  const int wg = blockIdx.x * 8 + wave;   // 0..383
  const int mi = wg & 7;                  // 8 M-tiles (one per wave in the block)
  const int nc = wg >> 3;                 // 0..47 N-chunks of 64 (one per block)
  const int row = mi * 16 + (lane & 15);
  const float* a_row;
  const float* W;
  float* Out;
  int colbase;
  if (nc < 24) {                          // gi: gather A row from embedding
    a_row = embed + (size_t)tok[row] * H_;
    W = W_ih; Out = GI; colbase = nc * 64;
  } else {                                // gh
    a_row = h + (size_t)row * H_;
    W = W_hh; Out = GH; colbase = (nc - 24) * 64;
  }
  v8f acc[4] = {};
  wave_tile(a_row, W, H_, colbase, lane, acc);
  const int half = lane >> 4, nl = lane & 15;
#pragma unroll
  for (int nn = 0; nn < 4; ++nn) {
    const int n = colbase + nn * 16 + nl;
#pragma unroll
    for (int j = 0; j < 8; ++j)
      Out[(size_t)(mi * 16 + j + 8 * half) * (3 * H_) + n] = acc[nn][j];
  }
}

// ---------------------------------------------------------------------------
// K2: per-batch-row: GRU gates -> h_new ; ONE-PASS streaming softmax-attention.
// mem tiles (16 x 512 f32 = 32KB) staged to LDS with double-buffered
// global_load_async_to_lds_b128 (ASYNCcnt).  Online max/sum rescaling keeps a
// single read of mem per step (~65 GB total L2 traffic instead of ~131 GB).
// grid = 128 blocks (one per b) x 512 thr (16 waves).
// ---------------------------------------------------------------------------
__global__ void k_attn(const float* __restrict__ GI, const float* __restrict__ GH,
                       const float* __restrict__ b_ih, const float* __restrict__ b_hh,
                       const float* __restrict__ mem, float* __restrict__ h,
                       float* __restrict__ CAT) {
  __shared__ float sh_tile[2][16 * H_];   // 2 x 32KB double buffer
  __shared__ float sh_h[H_];
  __shared__ float sh_sc[16];
  __shared__ float sh_w[16];
  const int b = blockIdx.x, tid = threadIdx.x;
  const int lane = tid & 31, wave = tid >> 5;
  const float* memb = mem + (size_t)b * LP_ * H_;
  const int NT = LP_ / 16;                // 32 tiles

  // --- prefetch tile 0 into buffer 0 (overlaps with gate math below) ---
  {
    const uint32_t lbase = (uint32_t)(uintptr_t)(&sh_tile[0][0]);
#pragma unroll
    for (int rnd = 0; rnd < 4; ++rnd)
      async_load_b128(lbase + rnd * 8192 + tid * 16,
                      (const void*)(memb + rnd * 2048 + tid * 4));
  }

  // --- GRU gates (one thread per hidden element) ---
  {
    const int j = tid;
    const float ir = GI[b * 1536 + j]        + b_ih[j];
    const float iz = GI[b * 1536 + 512 + j]  + b_ih[512 + j];
    const float in = GI[b * 1536 + 1024 + j] + b_ih[1024 + j];
    const float hr = GH[b * 1536 + j]        + b_hh[j];
    const float hz = GH[b * 1536 + 512 + j]  + b_hh[512 + j];
    const float hn = GH[b * 1536 + 1024 + j] + b_hh[1024 + j];
    const float r = 1.f / (1.f + expf(-(ir + hr)));
    const float z = 1.f / (1.f + expf(-(iz + hz)));
    const float n = tanhf(in + r * hn);
    const float hnew = (1.f - z) * n + z * h[b * H_ + j];
    h[b * H_ + j] = hnew;
    sh_h[j] = hnew;
    CAT[b * 1024 + j] = hnew;
  }
  __syncthreads();

  // --- streaming softmax-attention over 32 tiles of 16 rows ---
  float m_run = -INFINITY;   // replicated identically across all threads
  float s_run = 0.f;
  float ctx   = 0.f;         // thread owns hidden index `tid`

  for (int t = 0; t < NT; ++t) {
    const int buf = t & 1;
    if (t + 1 < NT) {        // prefetch next tile into other buffer
      const uint32_t lbase = (uint32_t)(uintptr_t)(&sh_tile[buf ^ 1][0]);
      const float* g = memb + (size_t)(t + 1) * 16 * H_;
#pragma unroll
      for (int rnd = 0; rnd < 4; ++rnd)
        async_load_b128(lbase + rnd * 8192 + tid * 16,
                        (const void*)(g + rnd * 2048 + tid * 4));
      WAIT_ASYNC(4);         // current tile's 4 issues retired; next stays in flight
    } else {
      WAIT_ASYNC(0);
    }
    __syncthreads();         // current tile visible to all waves

    // scores: wave w computes row w of the tile (wave32 shuffle reduction)
    {
      const float* row = &sh_tile[buf][wave * H_];
      float acc = 0.f;
#pragma unroll
      for (int i = 0; i < 16; ++i) acc += sh_h[lane + 32 * i] * row[lane + 32 * i];
      for (int off = 16; off; off >>= 1) acc += __shfl_xor(acc, off, 32);
      if (lane == 0) {
        const int l = t * 16 + wave;
        sh_sc[wave] = (l < L_) ? acc : -INFINITY;
      }
    }
    __syncthreads();

    // online max/sum update (replicated scalar state)
    float tmax = sh_sc[0];
#pragma unroll
    for (int r = 1; r < 16; ++r) tmax = fmaxf(tmax, sh_sc[r]);
    const float m_new = fmaxf(m_run, tmax);
    const float corr  = expf(m_run - m_new);   // 0 on first tile (m_run = -inf)
    s_run *= corr;
    ctx   *= corr;
    if (tid < 16) sh_w[tid] = expf(sh_sc[tid] - m_new);  // exp(-inf)=0 masks pad rows
    m_run = m_new;
    __syncthreads();

    // accumulate weighted rows into per-thread ctx (conflict-free LDS reads)
#pragma unroll
    for (int r = 0; r < 16; ++r) {
      const float w = sh_w[r];
      ctx   += w * sh_tile[buf][r * H_ + tid];
      s_run += w;
    }
    __syncthreads();         // protect buf before it is refilled next iteration
  }

  CAT[b * 1024 + 512 + tid] = ctx / s_run;
}

// ---------------------------------------------------------------------------
// K3: CC = tanh(CAT @ cat_W^T + cat_b)   (M=128, N=512, K=1024)
// grid = 8 blocks x 256 thr.
// ---------------------------------------------------------------------------
__global__ void k_cat(const float* __restrict__ CAT, const float* __restrict__ cat_W,
                      const float* __restrict__ cat_b, float* __restrict__ CC) {
  const int lane = threadIdx.x & 31, wave = threadIdx.x >> 5;
  const int mi = blockIdx.x;
  const int colbase = wave * 64;
  const int row = mi * 16 + (lane & 15);
  const float* a_row = CAT + (size_t)row * 1024;
  v8f acc[4] = {};
  wave_tile(a_row, cat_W, 1024, colbase, lane, acc);
  const int half = lane >> 4, nl = lane & 15;
#pragma unroll
  for (int nn = 0; nn < 4; ++nn) {
    const int n = colbase + nn * 16 + nl;
    const float bias = cat_b[n];
#pragma unroll
    for (int j = 0; j < 8; ++j)
      CC[(size_t)(mi * 16 + j + 8 * half) * H_ + n] = tanhf(acc[nn][j] + bias);
  }
}

// ---------------------------------------------------------------------------
// K4: logits = CC @ out_W^T + out_b (V=33); write out[b,v,t]; argmax -> tok.
// grid = 128 blocks x 128 thr (4 waves).
// ---------------------------------------------------------------------------
__global__ void k_out(const float* __restrict__ CC, const float* __restrict__ out_W,
                      const float* __restrict__ out_b, float* __restrict__ out,
                      int* __restrict__ tok, int t) {
  __shared__ float sh_cc[H_];
  __shared__ float sh_lg[V_];
  const int b = blockIdx.x, tid = threadIdx.x;
  for (int i = tid; i < H_; i += 128) sh_cc[i] = CC[b * H_ + i];
  __syncthreads();
  const int lane = tid & 31, wave = tid >> 5;
  for (int v = wave; v < V_; v += 4) {
    float acc = 0.f;
#pragma unroll
    for (int i = 0; i < 16; ++i)
      acc += sh_cc[lane + 32 * i] * out_W[v * H_ + lane + 32 * i];
    for (int off = 16; off; off >>= 1) acc += __shfl_xor(acc, off, 32);
    if (lane == 0) {
      const float lg = acc + out_b[v];
      sh_lg[v] = lg;
      out[((size_t)b * V_ + v) * L_ + t] = lg;
    }
  }
  __syncthreads();
  if (tid == 0) {   // first-max argmax (matches jnp.argmax tie-break)
    int best = 0; float bv = sh_lg[0];
    for (int v = 1; v < V_; ++v) if (sh_lg[v] > bv) { bv = sh_lg[v]; best = v; }
    tok[b] = best;
  }
}

// ---------------------------------------------------------------------------
extern "C" void kernel_launch(void* const* d_in, const int* in_sizes, int n_in,
                              void* d_out, int out_size, void* d_ws, size_t ws_size,
                              hipStream_t stream) {
  const float* latent = (const float*)d_in[0];   // [128,256]
  const float* up     = (const float*)d_in[1];   // [128,500,128]
  // d_in[2] = target (unused: eval-mode greedy feedback)
  const float* embed  = (const float*)d_in[3];   // [33,512]
  const float* hid_W  = (const float*)d_in[4];   // [512,256]
  const float* hid_b  = (const float*)d_in[5];   // [512]
  const float* mem_W  = (const float*)d_in[6];   // [512,128]
  const float* mem_b  = (const float*)d_in[7];   // [512]
  const float* W_ih   = (const float*)d_in[8];   // [1536,512]
  const float* W_hh   = (const float*)d_in[9];   // [1536,512]
  const float* b_ih   = (const float*)d_in[10];  // [1536]
  const float* b_hh   = (const float*)d_in[11];  // [1536]
  const float* cat_W  = (const float*)d_in[12];  // [512,1024]
  const float* cat_b  = (const float*)d_in[13];  // [512]
  const float* out_W  = (const float*)d_in[14];  // [33,512]
  const float* out_b  = (const float*)d_in[15];  // [33]
  float* out = (float*)d_out;                    // [128,33,500]

  // workspace carve-up (floats)
  float* ws  = (float*)d_ws;
  float* mem = ws;                                   // 128*512*512 (padded)
  float* h   = mem + (size_t)B_ * LP_ * H_;
  float* GI  = h   + (size_t)B_ * H_;
  float* GH  = GI  + (size_t)B_ * 3 * H_;
  float* CAT = GH  + (size_t)B_ * 3 * H_;
  float* CC  = CAT + (size_t)B_ * 2 * H_;
  int*   tok = (int*)(CC + (size_t)B_ * H_);

  // precompute attention memory (WMMA GEMM; L2-resident) + h0 + tok init
  k_mem<<<dim3(B_ * 32), dim3(256), 0, stream>>>(up, mem_W, mem_b, mem);
  k_h0 <<<dim3(8), dim3(256), 0, stream>>>(latent, hid_W, hid_b, h, tok);

  // 500 sequential decode steps (stream order enforces the dependency chain)
  for (int t = 0; t < L_; ++t) {
    k_gates_gemm<<<dim3(48),  dim3(256), 0, stream>>>(embed, h, tok, W_ih, W_hh, GI, GH);
    k_attn      <<<dim3(B_),  dim3(512), 0, stream>>>(GI, GH, b_ih, b_hh, mem, h, CAT);
    k_cat       <<<dim3(8),   dim3(256), 0, stream>>>(CAT, cat_W, cat_b, CC);
    k_out       <<<dim3(B_),  dim3(128), 0, stream>>>(CC, out_W, out_b, out, tok, t);
  }
}